// ModelNew_3556232921899
// MI455X (gfx1250) — compile-verified
//
#include <hip/hip_runtime.h>

// ---------------------------------------------------------------------------
// out[b] = ( x[b,:] . colsum(W) ) * 0.75,  x:16384x4096 f32, W:4096x4096 f32.
// Pure HBM-bound: ~320 MB reads -> ~13.7 us floor @ 23.3 TB/s.
//  K1: colsum via matrix pipe (ones x W with v_wmma_f32_16x16x4_f32),
//      W tiles staged to LDS by the Tensor Data Mover (double-buffered).
//  K2: deterministic slice-partial fold.
//  K3: streaming b128 GEMV, colsum in LDS, wave32 shuffle reduction.
// ---------------------------------------------------------------------------

typedef float        v2f __attribute__((ext_vector_type(2)));
typedef float        v8f __attribute__((ext_vector_type(8)));
typedef unsigned int v4u __attribute__((ext_vector_type(4)));
typedef int          v4i __attribute__((ext_vector_type(4)));
typedef int          v8i __attribute__((ext_vector_type(8)));

#define BATCH        16384
#define IN_SIZE      4096
#define HID_SIZE     4096
#define SLICES       16          // row-chunk partials (deterministic, no atomics)
#define COLS_PER_BLK 128         // 8 waves * 16 columns
#define STAGE_ROWS   64          // tile: 64 x 128 f32 = 32 KB
#define SCALE        0.75f       // SCALING_FACTOR / 2

#if __has_builtin(__builtin_amdgcn_wmma_f32_16x16x4_f32)
#define HAVE_WMMA_F32X4 1
#else
#define HAVE_WMMA_F32X4 0
#endif

#if defined(__gfx1250__) && __has_builtin(__builtin_amdgcn_tensor_load_to_lds) && \
    __has_builtin(__builtin_amdgcn_s_wait_tensorcnt)
#define HAVE_TDM 1
#else
#define HAVE_TDM 0
#endif

// therock-10.0 headers ship the 6-arg TDM builtin; ROCm 7.2 has the 5-arg one.
#if __has_include(<hip/amd_detail/amd_gfx1250_TDM.h>)
#define TDM_6ARG 1
#else
#define TDM_6ARG 0
#endif

// ---------------------------------------------------------------------------
// Kernel 1: partial column sums of W on the matrix pipe.
//   D(16x16) = ones(16x4) x Wchunk(4x16) + C : 4 rows x 16 cols per WMMA; with
//   A==ones the B K-permutation is irrelevant, only N=lane%16 matters.
//   Tiles reach LDS via TENSOR_LOAD_TO_LDS (TENSORcnt), double-buffered.
// grid = (IN/128, SLICES), block = 256 (8 wave32)
// ---------------------------------------------------------------------------
__global__ __launch_bounds__(256)
void k_colsum_wmma(const float* __restrict__ W, float* __restrict__ partial)
{
    __shared__ float tile[2][STAGE_ROWS * COLS_PER_BLK];   // 2 x 32 KB

    const int tid  = threadIdx.x;
    const int lane = tid & 31;
    const int wave = tid >> 5;
    const int k0   = blockIdx.x * COLS_PER_BLK;            // column base
    const int rows_per_slice = HID_SIZE / SLICES;          // 256
    const int j0   = blockIdx.y * rows_per_slice;          // row base
    const int nstages = rows_per_slice / STAGE_ROWS;       // 4

#if HAVE_WMMA_F32X4
    v8f acc = {0.f, 0.f, 0.f, 0.f, 0.f, 0.f, 0.f, 0.f};
    const v2f ones = {1.0f, 1.0f};
    const int wcol  = wave * 16 + (lane & 15);             // column inside tile
    const int half2 = (lane >> 4) * 2;                     // K sub-slot (perm-free)
#else
    float facc = 0.0f;
#endif

#if HAVE_TDM
    // ---- Tensor DMA descriptor (ISA 08 §8): 2D tensor, 4B elements --------
    const uint64_t wbase    = (uint64_t)(const void*)W;
    const uint32_t lds_base = (uint32_t)(uint64_t)(const void*)&tile[0][0];
    const v8i g1 = {
        (int)(2u << 16),                                // wg_mask=0, data_size=4B
        (int)(((uint32_t)IN_SIZE  & 0xFFFFu) << 16),    // tensor_dim0[15:0]<<16
        (int)(((uint32_t)HID_SIZE & 0xFFFFu) << 16),    // dim0 hi=0 | tensor_dim1 lo<<16
        (int)(((uint32_t)COLS_PER_BLK)       << 16),    // dim1 hi=0 | tile_dim0<<16
        (int)STAGE_ROWS,                                // tile_dim1 | tile_dim2=0
        (int)IN_SIZE,                                   // tensor_dim0_stride[31:0]
        0, 0                                            // stride hi, dim1_stride
    };
    const v4i z4 = {0, 0, 0, 0};
#if TDM_6ARG
    const v8i z8 = {0, 0, 0, 0, 0, 0, 0, 0};
#endif
    auto tdm_issue = [&](int buf, int row) {
        const uint64_t ga = wbase + 4ull * ((uint64_t)(j0 + row) * IN_SIZE + k0);
        v4u g0;
        g0.x = 1u;                                                   // count=1
        g0.y = lds_base + (uint32_t)buf * (STAGE_ROWS * COLS_PER_BLK * 4);
        g0.z = (uint32_t)ga;                                         // addr[31:0]
        g0.w = (uint32_t)((ga >> 32) & 0x01FFFFFFu) | (2u << 30);    // addr[56:32], type=2
#if TDM_6ARG
        __builtin_amdgcn_tensor_load_to_lds(g0, g1, z4, z4, z8, 0);
#else
        __builtin_amdgcn_tensor_load_to_lds(g0, g1, z4, z4, 0);
#endif
    };

    if (wave == 0) tdm_issue(0, 0);                 // prime the pipeline
#endif

    for (int s = 0; s < nstages; ++s) {
#if HAVE_TDM
        const int buf = s & 1;
        if (s + 1 < nstages) {
            if (wave == 0) tdm_issue((s + 1) & 1, (s + 1) * STAGE_ROWS);
            __builtin_amdgcn_s_wait_tensorcnt((short)1);   // current tile done
        } else {
            __builtin_amdgcn_s_wait_tensorcnt((short)0);
        }
        __syncthreads();          // wave0's completed wait dominates all reads
#else
        const int buf = 0;
        __syncthreads();          // previous tile fully consumed
        // Manual staging: 256 threads x 8 float4 (b128, row-coalesced).
        #pragma unroll
        for (int q = 0; q < 8; ++q) {
            const int f  = tid + q * 256;              // float4 index 0..2047
            const int r  = f >> 5;                     // tile row 0..63
            const int c4 = f & 31;                     // float4 col 0..31
            const float4* src = reinterpret_cast<const float4*>(
                W + (size_t)(j0 + s * STAGE_ROWS + r) * IN_SIZE + k0) + c4;
            reinterpret_cast<float4*>(tile[0])[f] = *src;
        }
        __syncthreads();
#endif

        const float* tb = tile[buf];
#if HAVE_WMMA_F32X4
        #pragma unroll
        for (int kk = 0; kk < STAGE_ROWS; kk += 4) {
            v2f b;
            b.x = tb[(kk + half2    ) * COLS_PER_BLK + wcol];
            b.y = tb[(kk + half2 + 1) * COLS_PER_BLK + wcol];
            acc = __builtin_amdgcn_wmma_f32_16x16x4_f32(
                    false, ones, false, b, (short)0, acc, false, false);
        }
#else
        if (tid < COLS_PER_BLK) {
            #pragma unroll 8
            for (int kk = 0; kk < STAGE_ROWS; ++kk)
                facc += tb[kk * COLS_PER_BLK + tid];
        }
#endif

#if HAVE_TDM
        __syncthreads();          // buffer free before wave0 re-issues into it
#endif
    }

#if HAVE_WMMA_F32X4
    // D row M=0 (lanes 0-15, VGPR0) holds per-column sums; lanes>=16 dup M=8.
    if (lane < 16)
        partial[(size_t)blockIdx.y * IN_SIZE + k0 + wave * 16 + lane] = acc[0];
#else
    if (tid < COLS_PER_BLK)
        partial[(size_t)blockIdx.y * IN_SIZE + k0 + tid] = facc;
#endif
}

// ---------------------------------------------------------------------------
// Kernel 2: fold 16 deterministic slice-partials -> final colsum (16 KB).
// ---------------------------------------------------------------------------
__global__ __launch_bounds__(256)
void k_reduce_partials(const float* __restrict__ partial,
                       float* __restrict__ colsum)
{
    const int k = blockIdx.x * 256 + threadIdx.x;          // 0..4095
    float s = 0.0f;
    #pragma unroll
    for (int c = 0; c < SLICES; ++c)
        s += partial[(size_t)c * IN_SIZE + k];
    colsum[k] = s;
}

// ---------------------------------------------------------------------------
// Kernel 3: out[row] = 0.75 * dot(x[row,:], colsum).  One wave32 per row,
// b128 streaming of x (the 256 MB term), colsum LDS-resident (L2-hot),
// global_prefetch_b8 one 1 KB step ahead, wave32 shuffle tree reduce.
// grid = 2048, block = 256
// ---------------------------------------------------------------------------
__global__ __launch_bounds__(256)
void k_gemv_colsum(const float* __restrict__ x,
                   const float* __restrict__ colsum,
                   float* __restrict__ out)
{
    __shared__ float s_lds[IN_SIZE];                       // 16 KB

    const int tid = threadIdx.x;
    #pragma unroll
    for (int j = 0; j < IN_SIZE / 4 / 256; ++j)            // 4 x b128 per thread
        reinterpret_cast<float4*>(s_lds)[tid + j * 256] =
            reinterpret_cast<const float4*>(colsum)[tid + j * 256];
    __syncthreads();

    const int lane = tid & 31;
    const int wave = tid >> 5;
    const int row  = blockIdx.x * 8 + wave;

    const float4* xr  = reinterpret_cast<const float4*>(x + (size_t)row * IN_SIZE);
    const float4* sv4 = reinterpret_cast<const float4*>(s_lds);

    float4 a = {0.f, 0.f, 0.f, 0.f};
    for (int i = lane; i < IN_SIZE / 4; i += 32) {         // 32 iters, b128 each
        __builtin_prefetch(xr + i + 64, 0, 0);             // speculative, +1 KB
        const float4 xv = xr[i];
        const float4 sv = sv4[i];
        a.x += xv.x * sv.x;
        a.y += xv.y * sv.y;
        a.z += xv.z * sv.z;
        a.w += xv.w * sv.w;
    }
    float r = (a.x + a.y) + (a.z + a.w);
    #pragma unroll
    for (int off = 16; off > 0; off >>= 1)                 // wave32 tree reduce
        r += __shfl_xor(r, off, 32);

    if (lane == 0)
        out[row] = r * SCALE;
}

// ---------------------------------------------------------------------------
extern "C" void kernel_launch(void* const* d_in, const int* in_sizes, int n_in,
                              void* d_out, int out_size, void* d_ws, size_t ws_size,
                              hipStream_t stream)
{
    const float* x = (const float*)d_in[0];                // (16384, 4096) f32
    const float* W = (const float*)d_in[1];                // (4096, 4096)  f32
    float* out     = (float*)d_out;                        // 16384 f32

    float* partial = (float*)d_ws;                         // SLICES*4096 f32
    float* colsum  = partial + (size_t)SLICES * IN_SIZE;   // 4096 f32

    k_colsum_wmma<<<dim3(IN_SIZE / COLS_PER_BLK, SLICES), 256, 0, stream>>>(W, partial);
    k_reduce_partials<<<IN_SIZE / 256, 256, 0, stream>>>(partial, colsum);
    k_gemv_colsum<<<BATCH / 8, 256, 0, stream>>>(x, colsum, out);
}